// Memory_39170101739965
// MI455X (gfx1250) — compile-verified
//
#include <hip/hip_runtime.h>
#include <hip/hip_bf16.h>

typedef __attribute__((ext_vector_type(2))) float v2f;
typedef __attribute__((ext_vector_type(8))) float v8f;
typedef int v4i __attribute__((vector_size(4 * sizeof(int))));
typedef __attribute__((address_space(1))) v4i gv4i;   // global int4
typedef __attribute__((address_space(3))) v4i lv4i;   // LDS int4

#define B_SZ 2048
#define N_SZ 512
#define U_SZ 64
#define RSTR 68            // padded LDS row stride (floats): 16B aligned, bank-spreading
#define NTHREADS 256

// LDS layout (float indices)
#define LDS_M    0
#define LDS_SIM  (N_SZ * RSTR)          // 34816 : raw dot products
#define LDS_W    (LDS_SIM + N_SZ)       // gated weights wg
#define LDS_WS   (LDS_W + N_SZ)         // final weights w
#define LDS_K    (LDS_WS + N_SZ)        // k vector (64)
#define LDS_E    (LDS_K + U_SZ)         // erase (64)
#define LDS_A    (LDS_E + U_SZ)         // add (64)
#define LDS_NRM  (LDS_A + U_SZ)         // row norms (512)
#define LDS_RED  (LDS_NRM + N_SZ)       // reduction scratch (16)
#define LDS_SCAL (LDS_RED + 16)         // scalars (8)
#define LDS_TOT  (LDS_SCAL + 8)         // 37080 floats = 148320 B -> 2 blocks/WGP

#define OUT_W_OFF  0
#define OUT_R_OFF  (B_SZ * N_SZ)                 // 1048576
#define OUT_M_OFF  (B_SZ * N_SZ + B_SZ * U_SZ)   // 1179648

#if defined(__has_builtin)
#  if __has_builtin(__builtin_amdgcn_global_load_async_to_lds_b128)
#    define HAVE_ASYNC_LDS 1
#  else
#    define HAVE_ASYNC_LDS 0
#  endif
#else
#  define HAVE_ASYNC_LDS 0
#endif

__device__ __forceinline__ float blockReduceSum(float v, float* red) {
    #pragma unroll
    for (int off = 16; off; off >>= 1) v += __shfl_xor(v, off, 32);
    const int wave = threadIdx.x >> 5;
    const int lane = threadIdx.x & 31;
    if (lane == 0) red[wave] = v;
    __syncthreads();
    float s = 0.f;
    #pragma unroll
    for (int i = 0; i < 8; ++i) s += red[i];
    __syncthreads();
    return s;
}

__global__ __launch_bounds__(NTHREADS)
void ntm_head_kernel(const float* __restrict__ mem,
                     const float* __restrict__ kvec,
                     const float* __restrict__ beta_p,
                     const float* __restrict__ g_p,
                     const float* __restrict__ s_p,
                     const float* __restrict__ gamma_p,
                     const float* __restrict__ wpre,
                     const float* __restrict__ e_p,
                     const float* __restrict__ a_p,
                     float* __restrict__ out) {
    extern __shared__ float lds[];
    const int t = threadIdx.x;
    const int b = blockIdx.x;

    float* Mt   = lds + LDS_M;
    float* simb = lds + LDS_SIM;
    float* wb   = lds + LDS_W;
    float* wsb  = lds + LDS_WS;
    float* kb   = lds + LDS_K;
    float* eb   = lds + LDS_E;
    float* ab   = lds + LDS_A;
    float* nb   = lds + LDS_NRM;
    float* red  = lds + LDS_RED;
    float* scal = lds + LDS_SCAL;

    // ---------------- Phase 0: stage memory[b] (128KB) + small vectors into LDS
#if HAVE_ASYNC_LDS
    {
        // Direct global->LDS DMA path (ASYNCcnt), no VGPR round-trip: each lane
        // issues 32 b128 async copies back-to-back for max memory-level parallelism.
        const float* gbase = mem + (size_t)b * (N_SZ * U_SZ);
        #pragma unroll
        for (int i = 0; i < 32; ++i) {
            int c = i * NTHREADS + t;                 // float4 chunk id, coalesced
            int row = c >> 4, seg = c & 15;
            gv4i* gp = (gv4i*)(uintptr_t)(gbase + (size_t)c * 4);
            lv4i* lp = (lv4i*)(uint32_t)(uintptr_t)(&Mt[row * RSTR + seg * 4]);
            __builtin_amdgcn_global_load_async_to_lds_b128(gp, lp, 0, 0);
        }
    }
#else
    {
        const float4* gm = reinterpret_cast<const float4*>(mem) + (size_t)b * (N_SZ * U_SZ / 4);
        #pragma unroll 8
        for (int i = 0; i < 32; ++i) {
            int c = i * NTHREADS + t;
            float4 v = gm[c];
            int row = c >> 4, seg = c & 15;
            *reinterpret_cast<float4*>(&Mt[row * RSTR + seg * 4]) = v;
        }
    }
#endif
    if (t < U_SZ) {
        kb[t] = kvec[b * U_SZ + t];
        eb[t] = e_p[b * U_SZ + t];
        ab[t] = a_p[b * U_SZ + t];
    }
    if (t == 0) {
        scal[0] = beta_p[0];
        scal[1] = g_p[0];
        scal[2] = gamma_p[0];
        scal[3] = s_p[3 * b + 0];
        scal[4] = s_p[3 * b + 1];
        scal[5] = s_p[3 * b + 2];
    }
    __builtin_prefetch(&wpre[b * N_SZ + t], 0, 1);   // global_prefetch_b8
#if HAVE_ASYNC_LDS
#  if __has_builtin(__builtin_amdgcn_s_wait_asynccnt)
    __builtin_amdgcn_s_wait_asynccnt(0);             // our async copies are LDS-visible
#  else
    asm volatile("s_wait_asynccnt 0" ::: "memory");
#  endif
#endif
    __syncthreads();

    // ---------------- Phase 1: row norms (staggered, bank-conflict-free) + ||k||
    {
        float acc0 = 0.f, acc1 = 0.f;
        #pragma unroll 8
        for (int i = 0; i < U_SZ; ++i) {
            int idx = (i + t) & 63;
            float x0 = Mt[t * RSTR + idx];
            float x1 = Mt[(t + 256) * RSTR + idx];
            acc0 = fmaf(x0, x0, acc0);
            acc1 = fmaf(x1, x1, acc1);
        }
        nb[t]       = sqrtf(acc0);
        nb[t + 256] = sqrtf(acc1);
    }
    float kn2 = 0.f;
    #pragma unroll 8
    for (int i = 0; i < U_SZ; ++i) kn2 = fmaf(kb[i], kb[i], kn2);  // broadcast reads
    const float knorm = fmaxf(sqrtf(kn2), 1e-8f);

    // ---------------- Phase 2: dot(memory_row, k) via V_WMMA_F32_16X16X4_F32 (exact f32)
    // A = k chunk replicated on all 16 rows, B = 16 memory rows as columns.
    // D[m,n] = sim[16*grp + n] for every m  ->  layout-robust extraction.
    {
        const int wv = t >> 5;
        const int L = t & 31;
        const int half = L >> 4;
        const int lm = L & 15;
        #pragma unroll
        for (int gg = 0; gg < 4; ++gg) {
            const int grp = wv + 8 * gg;             // 0..31 row-groups of 16
            const int rb = (16 * grp + lm) * RSTR;
            v8f acc = {0.f, 0.f, 0.f, 0.f, 0.f, 0.f, 0.f, 0.f};
            #pragma unroll
            for (int kc = 0; kc < 16; ++kc) {        // K = 4 per WMMA, U = 64
                const int ku = 4 * kc + 2 * half;
                v2f Av, Bv;
                Av[0] = kb[ku];
                Av[1] = kb[ku + 1];
                Bv[0] = Mt[rb + ku];
                Bv[1] = Mt[rb + ku + 1];
                acc = __builtin_amdgcn_wmma_f32_16x16x4_f32(
                    false, Av, false, Bv, (short)0, acc, false, false);
            }
            if (L < 16) simb[16 * grp + L] = acc[0];
        }
    }
    __syncthreads();

    // ---------------- Phase 3: softmax(beta*cos) -> gate -> circular shift -> sharpen
    const float be = scal[0], gv = scal[1], ga = scal[2];
    const float s0 = scal[3], s1 = scal[4], s2 = scal[5];
    float ev0, ev1;
    {
        float c0 = simb[t] / (fmaxf(nb[t], 1e-8f) * knorm);
        float c1 = simb[t + 256] / (fmaxf(nb[t + 256], 1e-8f) * knorm);
        ev0 = expf(be * c0);          // |be*c| <= beta, safe without max-subtract
        ev1 = expf(be * c1);
    }
    const float sumE = blockReduceSum(ev0 + ev1, red);
    {
        float wp0 = wpre[b * N_SZ + t];
        float wp1 = wpre[b * N_SZ + t + 256];
        wb[t]       = gv * (ev0 / sumE) + (1.f - gv) * wp0;
        wb[t + 256] = gv * (ev1 / sumE) + (1.f - gv) * wp1;
    }
    __syncthreads();
    float sh0, sh1;
    {
        int n0 = t, n1 = t + 256;
        float ws0 = s0 * wb[(n0 + 511) & 511] + s1 * wb[n0] + s2 * wb[(n0 + 1) & 511];
        float ws1 = s0 * wb[(n1 + 511) & 511] + s1 * wb[n1] + s2 * wb[(n1 + 1) & 511];
        sh0 = powf(ws0, ga);
        sh1 = powf(ws1, ga);
    }
    const float sumS = blockReduceSum(sh0 + sh1, red);
    {
        float w0 = sh0 / sumS + 1e-16f;
        float w1 = sh1 / sumS + 1e-16f;
        wsb[t] = w0;
        wsb[t + 256] = w1;
        out[OUT_W_OFF + (size_t)b * N_SZ + t] = w0;
        out[OUT_W_OFF + (size_t)b * N_SZ + t + 256] = w1;
    }
    __syncthreads();

    // ---------------- Phase 4: r[u] = sum_n w[n] * M[n][u]
    {
        const int u = t & 63;
        const int grp = t >> 6;                      // 4 partial groups
        float acc = 0.f;
        #pragma unroll 8
        for (int j = 0; j < 128; ++j) {
            int n = 4 * j + grp;
            acc = fmaf(wsb[n], Mt[n * RSTR + u], acc);
        }
        simb[t] = acc;                               // reuse simb as scratch
        __syncthreads();
        if (t < U_SZ) {
            out[OUT_R_OFF + (size_t)b * U_SZ + t] =
                simb[t] + simb[t + 64] + simb[t + 128] + simb[t + 192];
        }
    }

    // ---------------- Phase 5: new_mem = M*(1 - w e^T) + w a^T  (coalesced float4)
    {
        float4* gout = reinterpret_cast<float4*>(out + OUT_M_OFF) + (size_t)b * (N_SZ * U_SZ / 4);
        #pragma unroll 4
        for (int i = 0; i < 32; ++i) {
            int c = i * NTHREADS + t;
            int row = c >> 4, seg = c & 15, u0 = seg * 4;
            float4 m = *reinterpret_cast<const float4*>(&Mt[row * RSTR + u0]);
            float wn = wsb[row];
            float4 o;
            o.x = m.x * (1.f - wn * eb[u0 + 0]) + wn * ab[u0 + 0];
            o.y = m.y * (1.f - wn * eb[u0 + 1]) + wn * ab[u0 + 1];
            o.z = m.z * (1.f - wn * eb[u0 + 2]) + wn * ab[u0 + 2];
            o.w = m.w * (1.f - wn * eb[u0 + 3]) + wn * ab[u0 + 3];
            gout[c] = o;
        }
    }
}

extern "C" void kernel_launch(void* const* d_in, const int* in_sizes, int n_in,
                              void* d_out, int out_size, void* d_ws, size_t ws_size,
                              hipStream_t stream) {
    (void)in_sizes; (void)n_in; (void)out_size; (void)d_ws; (void)ws_size;
    const float* mem   = (const float*)d_in[0];
    const float* k     = (const float*)d_in[1];
    const float* beta  = (const float*)d_in[2];
    const float* g     = (const float*)d_in[3];
    const float* s     = (const float*)d_in[4];
    const float* gamma = (const float*)d_in[5];
    const float* wpre  = (const float*)d_in[6];
    const float* e     = (const float*)d_in[7];
    const float* a     = (const float*)d_in[8];
    float* out = (float*)d_out;

    const size_t shmem = (size_t)LDS_TOT * sizeof(float);   // ~148 KB dynamic LDS
    (void)hipFuncSetAttribute(reinterpret_cast<const void*>(ntm_head_kernel),
                              hipFuncAttributeMaxDynamicSharedMemorySize, (int)shmem);
    hipLaunchKernelGGL(ntm_head_kernel, dim3(B_SZ), dim3(NTHREADS), shmem, stream,
                       mem, k, beta, g, s, gamma, wpre, e, a, out);
}